// GCNEncoder_69329362092400
// MI455X (gfx1250) — compile-verified
//
#include <hip/hip_runtime.h>

typedef __attribute__((ext_vector_type(2))) float v2f;
typedef __attribute__((ext_vector_type(8))) float v8f;

// ---------------------------------------------------------------------------
// Utility kernels
// ---------------------------------------------------------------------------
__global__ void k_fill(float* __restrict__ p, float v, int n) {
  int i = blockIdx.x * blockDim.x + threadIdx.x;
  if (i < n) p[i] = v;
}

__global__ void k_count_deg(const long long* __restrict__ dst,
                            float* __restrict__ deg, int E) {
  int e = blockIdx.x * blockDim.x + threadIdx.x;
  if (e < E) atomicAdd(&deg[(int)dst[e]], 1.0f);
}

__global__ void k_rsqrt(const float* __restrict__ deg,
                        float* __restrict__ dis, int n) {
  int i = blockIdx.x * blockDim.x + threadIdx.x;
  if (i < n) dis[i] = rsqrtf(fmaxf(deg[i], 1.0f));
}

// ---------------------------------------------------------------------------
// FP32 WMMA GEMM: C[M,N] = A[M,K] @ B[K,N], all row-major.
// One wave (32 lanes) per 16x16 output tile, V_WMMA_F32_16X16X4_F32 over K.
// Requires M%16==0, N%16==0, K%4==0 (true here: M=100000, N in {64,32},
// K in {128,64}). Branch-out condition is wave-uniform, so EXEC is all-1s
// for every WMMA as the ISA requires.
// ---------------------------------------------------------------------------
__global__ void k_gemm_wmma_f32(const float* __restrict__ A,
                                const float* __restrict__ B,
                                float* __restrict__ C,
                                int M, int N, int K) {
  const int lane = threadIdx.x & 31;
  const int wave = threadIdx.x >> 5;
  const int tilesN = N >> 4;
  const int totalTiles = (M >> 4) * tilesN;
  const int tile = blockIdx.x * (blockDim.x >> 5) + wave;
  if (tile >= totalTiles) return;

  const int tm = tile / tilesN;
  const int tn = tile % tilesN;

  // A 16x4 layout: lane holds row m=lane&15, k = 2*(lane>>4) + vgpr
  // B 4x16 layout: lane holds col n=lane&15, k = 2*(lane>>4) + vgpr
  const int m    = lane & 15;
  const int koff = (lane >> 4) << 1;          // 0 or 2
  const float* Arow = A + (size_t)(tm * 16 + m) * K;
  const float* Bcol = B + (tn * 16 + m);      // m == n index for B lanes

  v8f c = {0.f, 0.f, 0.f, 0.f, 0.f, 0.f, 0.f, 0.f};

  for (int kb = 0; kb < K; kb += 4) {
    v2f a, b;
    a.x = Arow[kb + koff];
    a.y = Arow[kb + koff + 1];
    b.x = Bcol[(size_t)(kb + koff) * N];
    b.y = Bcol[(size_t)(kb + koff + 1) * N];
    // 8 args: (neg_a, A, neg_b, B, c_mod, C, reuse_a, reuse_b)
    c = __builtin_amdgcn_wmma_f32_16x16x4_f32(false, a, false, b,
                                              (short)0, c, false, false);
  }

  // C/D 16x16 f32: vgpr r, lane L -> row = 8*(L>>4)+r, col = L&15
  const int rbase = tm * 16 + ((lane >> 4) << 3);
  const int cc    = tn * 16 + (lane & 15);
  #pragma unroll
  for (int r = 0; r < 8; ++r)
    C[(size_t)(rbase + r) * N + cc] = c[r];
}

// ---------------------------------------------------------------------------
// Edge scatter: agg[dst] += dis[src]*dis[dst] * hw[src]   (float4 per thread)
// ---------------------------------------------------------------------------
__global__ void k_scatter(const float* __restrict__ hw,
                          float* __restrict__ agg,
                          const long long* __restrict__ src,
                          const long long* __restrict__ dst,
                          const float* __restrict__ dis,
                          int E, int F) {
  const int chunks = F >> 2;                       // float4 chunks per edge
  int gid = blockIdx.x * blockDim.x + threadIdx.x;
  int e = gid / chunks;
  int cch = gid - e * chunks;
  if (e >= E) return;
  int s = (int)src[e];
  int d = (int)dst[e];
  float nrm = dis[s] * dis[d];
  float4 v = reinterpret_cast<const float4*>(hw + (size_t)s * F)[cch];
  float* out = agg + (size_t)d * F + cch * 4;
  atomicAdd(out + 0, v.x * nrm);
  atomicAdd(out + 1, v.y * nrm);
  atomicAdd(out + 2, v.z * nrm);
  atomicAdd(out + 3, v.w * nrm);
}

// ---------------------------------------------------------------------------
// Finalize: out = [relu]( agg + dis[i]^2 * hw + bias )   (self-loop term)
// ---------------------------------------------------------------------------
__global__ void k_finalize(const float* __restrict__ agg,
                           const float* __restrict__ hw,
                           const float* __restrict__ dis,
                           const float* __restrict__ bias,
                           float* __restrict__ out,
                           int total, int F, int do_relu) {
  int gid = blockIdx.x * blockDim.x + threadIdx.x;
  if (gid >= total) return;
  int i = gid / F;
  int f = gid - i * F;
  float di = dis[i];
  float v = agg[gid] + di * di * hw[gid] + bias[f];
  out[gid] = do_relu ? fmaxf(v, 0.0f) : v;
}

// ---------------------------------------------------------------------------
// Launch
// ---------------------------------------------------------------------------
extern "C" void kernel_launch(void* const* d_in, const int* in_sizes, int n_in,
                              void* d_out, int out_size, void* d_ws, size_t ws_size,
                              hipStream_t stream) {
  const int FIN = 128, H = 64, O = 32;
  const int Nn = in_sizes[0] / FIN;     // 100000
  const int E  = in_sizes[1] / 2;       // 1600000

  const float*     x   = (const float*)d_in[0];
  const long long* ei  = (const long long*)d_in[1];
  const float*     W1  = (const float*)d_in[2];
  const float*     b1  = (const float*)d_in[3];
  const float*     Wmu = (const float*)d_in[4];
  const float*     bmu = (const float*)d_in[5];
  const float*     Wls = (const float*)d_in[6];
  const float*     bls = (const float*)d_in[7];
  float*           out = (float*)d_out;

  const long long* srcI = ei;         // edge_index[0]
  const long long* dstI = ei + E;     // edge_index[1]

  // Workspace layout (floats): deg[Nn] | dis[Nn] | bufA[Nn*64] | bufB[Nn*64] | bufC[Nn*64]
  float* deg  = (float*)d_ws;
  float* dis  = deg + Nn;
  float* bufA = dis + Nn;
  float* bufB = bufA + (size_t)Nn * H;
  float* bufC = bufB + (size_t)Nn * H;

  const int TB = 256;
  auto cdiv = [](long long a, long long b) { return (int)((a + b - 1) / b); };

  // 1) symmetric-norm coefficients
  k_fill<<<cdiv(Nn, TB), TB, 0, stream>>>(deg, 1.0f, Nn);        // self-loops
  k_count_deg<<<cdiv(E, TB), TB, 0, stream>>>(dstI, deg, E);
  k_rsqrt<<<cdiv(Nn, TB), TB, 0, stream>>>(deg, dis, Nn);

  // 2) layer 1: hw1 = x @ W1   (WMMA f32)
  {
    int tiles = (Nn / 16) * (H / 16);
    k_gemm_wmma_f32<<<cdiv(tiles, TB / 32), TB, 0, stream>>>(x, W1, bufA, Nn, H, FIN);
  }
  // 3) aggregate + relu -> h (in place in bufB)
  k_fill<<<cdiv((long long)Nn * H, TB), TB, 0, stream>>>(bufB, 0.0f, Nn * H);
  k_scatter<<<cdiv((long long)E * (H / 4), TB), TB, 0, stream>>>(bufA, bufB, srcI, dstI, dis, E, H);
  k_finalize<<<cdiv((long long)Nn * H, TB), TB, 0, stream>>>(bufB, bufA, dis, b1, bufB, Nn * H, H, 1);

  // 4) layer 2: hw_mu / hw_logstd = h @ W   (WMMA f32)
  {
    int tiles = (Nn / 16) * (O / 16);
    k_gemm_wmma_f32<<<cdiv(tiles, TB / 32), TB, 0, stream>>>(bufB, Wmu, bufA, Nn, O, H);
    k_gemm_wmma_f32<<<cdiv(tiles, TB / 32), TB, 0, stream>>>(bufB, Wls, bufA + (size_t)Nn * O, Nn, O, H);
  }
  // 5) aggregate both heads
  k_fill<<<cdiv((long long)Nn * 2 * O, TB), TB, 0, stream>>>(bufC, 0.0f, Nn * 2 * O);
  k_scatter<<<cdiv((long long)E * (O / 4), TB), TB, 0, stream>>>(bufA, bufC, srcI, dstI, dis, E, O);
  k_scatter<<<cdiv((long long)E * (O / 4), TB), TB, 0, stream>>>(bufA + (size_t)Nn * O, bufC + (size_t)Nn * O,
                                                                 srcI, dstI, dis, E, O);
  // 6) finalize straight into d_out: mu | logstd
  k_finalize<<<cdiv((long long)Nn * O, TB), TB, 0, stream>>>(bufC, bufA, dis, bmu, out, Nn * O, O, 0);
  k_finalize<<<cdiv((long long)Nn * O, TB), TB, 0, stream>>>(bufC + (size_t)Nn * O, bufA + (size_t)Nn * O,
                                                             dis, bls, out + (size_t)Nn * O, Nn * O, O, 0);
}